// TripletLoss_54846732370349
// MI455X (gfx1250) — compile-verified
//
#include <hip/hip_runtime.h>
#include <math.h>

typedef __attribute__((ext_vector_type(2))) float v2f;
typedef __attribute__((ext_vector_type(8))) float v8f;

#define TL_D      128
#define TL_MARGIN 0.3f

// ---------------------------------------------------------------------------
// Kernel 1: L2-normalize rows (torch F.normalize: x / max(||x||, eps)).
// One wave (32 lanes) per row; 4 floats/lane; wave32 shuffle-tree reduction.
// Also stores ||e||^2 of the normalized row (reference uses diag of the gram).
// ---------------------------------------------------------------------------
__global__ __launch_bounds__(256)
void tl_normalize(const float* __restrict__ x,
                  float* __restrict__ e,
                  float* __restrict__ sq) {
  const int row  = blockIdx.x * (blockDim.x >> 5) + (threadIdx.x >> 5);
  const int lane = threadIdx.x & 31;

  float4 v = ((const float4*)(x + (size_t)row * TL_D))[lane];
  float s = v.x * v.x + v.y * v.y + v.z * v.z + v.w * v.w;
#pragma unroll
  for (int off = 16; off > 0; off >>= 1) s += __shfl_xor(s, off, 32);

  const float inv = 1.0f / fmaxf(sqrtf(s), 1e-12f);
  v.x *= inv; v.y *= inv; v.z *= inv; v.w *= inv;
  ((float4*)(e + (size_t)row * TL_D))[lane] = v;

  float s2 = v.x * v.x + v.y * v.y + v.z * v.z + v.w * v.w;
#pragma unroll
  for (int off = 16; off > 0; off >>= 1) s2 += __shfl_xor(s2, off, 32);
  if (lane == 0) sq[row] = s2;
}

// ---------------------------------------------------------------------------
// Kernel 2: fused Gram (fp32 WMMA) + batch-hard mining.
// Block = 256 threads = 8 waves. Wave w owns rows [(blk*8+w)*16, +16).
// Each outer iteration handles TWO 16-column tiles, each accumulated with
// TWO K-split accumulators -> 4 independent v_wmma chains of length 16
// (hides WMMA latency at 2 waves/SIMD).
// C/D layout: VGPR r, lane l -> (M = r + 8*(l/16), N = l%16).
// ---------------------------------------------------------------------------
__global__ __launch_bounds__(256)
void tl_mine(const float* __restrict__ E,
             const float* __restrict__ sq,
             const int* __restrict__ labels,
             float* __restrict__ row_loss,
             float* __restrict__ row_valid,
             int B) {
  const int wave = threadIdx.x >> 5;
  const int lane = threadIdx.x & 31;
  const int half = lane >> 4;   // 0: rows 0-7 & K pair {0,1}; 1: rows 8-15 & K pair {2,3}
  const int l16  = lane & 15;
  const int row_base = (blockIdx.x * 8 + wave) * 16;
  const int rbase    = row_base + half * 8;   // first of this lane's 8 rows

  float sqr[8]; int labr[8];
#pragma unroll
  for (int r = 0; r < 8; ++r) {
    sqr[r]  = sq[rbase + r];
    labr[r] = labels[rbase + r];
  }

  float ap[8], an[8];
#pragma unroll
  for (int r = 0; r < 8; ++r) {
    ap[r] = -__builtin_inff();
    an[r] =  __builtin_inff();
  }

  // A fragment (16x4 f32, K-slice kk): lane l -> E[row_base+l%16][4*kk + 2*half + {0,1}]
  v2f afrag[32];
  {
    const float* arow = E + (size_t)(row_base + l16) * TL_D + 2 * half;
#pragma unroll
    for (int kk = 0; kk < 32; ++kk) afrag[kk] = *(const v2f*)(arow + 4 * kk);
  }

  for (int cb = 0; cb < B; cb += 32) {
    const int j0 = cb + l16;        // tile-0 column for this lane
    const int j1 = j0 + 16;         // tile-1 column
    const float* b0 = E + (size_t)j0 * TL_D + 2 * half;
    const float* b1 = E + (size_t)j1 * TL_D + 2 * half;

    v8f a0 = {0.f, 0.f, 0.f, 0.f, 0.f, 0.f, 0.f, 0.f};  // tile 0, even K slices
    v8f a1 = {0.f, 0.f, 0.f, 0.f, 0.f, 0.f, 0.f, 0.f};  // tile 0, odd  K slices
    v8f a2 = {0.f, 0.f, 0.f, 0.f, 0.f, 0.f, 0.f, 0.f};  // tile 1, even K slices
    v8f a3 = {0.f, 0.f, 0.f, 0.f, 0.f, 0.f, 0.f, 0.f};  // tile 1, odd  K slices

#pragma unroll
    for (int kk = 0; kk < 32; kk += 2) {
      v2f bf0a = *(const v2f*)(b0 + 4 * kk);
      v2f bf0b = *(const v2f*)(b0 + 4 * kk + 4);
      v2f bf1a = *(const v2f*)(b1 + 4 * kk);
      v2f bf1b = *(const v2f*)(b1 + 4 * kk + 4);
      a0 = __builtin_amdgcn_wmma_f32_16x16x4_f32(
          false, afrag[kk],     false, bf0a, (short)0, a0, false, false);
      a1 = __builtin_amdgcn_wmma_f32_16x16x4_f32(
          false, afrag[kk + 1], false, bf0b, (short)0, a1, false, false);
      a2 = __builtin_amdgcn_wmma_f32_16x16x4_f32(
          false, afrag[kk],     false, bf1a, (short)0, a2, false, false);
      a3 = __builtin_amdgcn_wmma_f32_16x16x4_f32(
          false, afrag[kk + 1], false, bf1b, (short)0, a3, false, false);
    }

    const float sq0 = sq[j0];
    const float sq1 = sq[j1];
    const int   lb0 = labels[j0];
    const int   lb1 = labels[j1];
    const int   dr0 = j0 - rbase;   // diagonal for tile 0 when dr0 == r
    const int   dr1 = j1 - rbase;   // diagonal for tile 1 when dr1 == r

#pragma unroll
    for (int r = 0; r < 8; ++r) {
      const float d20 = sqr[r] + sq0 - 2.0f * (a0[r] + a1[r]);
      const float d21 = sqr[r] + sq1 - 2.0f * (a2[r] + a3[r]);
      if (labr[r] == lb0) {
        if (dr0 != r) ap[r] = fmaxf(ap[r], d20);   // hardest positive
      } else {
        an[r] = fminf(an[r], d20);                 // hardest negative
      }
      if (labr[r] == lb1) {
        if (dr1 != r) ap[r] = fmaxf(ap[r], d21);
      } else {
        an[r] = fminf(an[r], d21);
      }
    }
  }

  // Reduce each row's candidates across its 16 lanes (wave32 shuffle tree).
#pragma unroll
  for (int r = 0; r < 8; ++r) {
#pragma unroll
    for (int off = 8; off > 0; off >>= 1) {
      ap[r] = fmaxf(ap[r], __shfl_xor(ap[r], off, 16));
      an[r] = fminf(an[r], __shfl_xor(an[r], off, 16));
    }
  }

  if (l16 == 0) {
#pragma unroll
    for (int r = 0; r < 8; ++r) {
      const int   row   = rbase + r;
      const bool  valid = (ap[r] > -1e30f) && (an[r] < 1e30f);
      const float dap   = sqrtf(fmaxf(ap[r], 0.0f));   // sqrt monotone:
      const float dan   = sqrtf(fmaxf(an[r], 0.0f));   // argext on d2 == on dist
      const float pl    = fmaxf(dap - dan + TL_MARGIN, 0.0f);
      row_loss[row]  = valid ? pl : 0.0f;
      row_valid[row] = valid ? 1.0f : 0.0f;
    }
  }
}

// ---------------------------------------------------------------------------
// Kernel 3: deterministic finalize (fixed-shape tree reduction, no atomics).
// ---------------------------------------------------------------------------
__global__ __launch_bounds__(256)
void tl_finalize(const float* __restrict__ row_loss,
                 const float* __restrict__ row_valid,
                 float* __restrict__ out, int B) {
  __shared__ float sl[256];
  __shared__ float sv[256];
  const int t = threadIdx.x;
  float a = 0.0f, b = 0.0f;
  for (int i = t; i < B; i += 256) { a += row_loss[i]; b += row_valid[i]; }
  sl[t] = a; sv[t] = b;
  __syncthreads();
  for (int s = 128; s > 0; s >>= 1) {
    if (t < s) { sl[t] += sl[t + s]; sv[t] += sv[t + s]; }
    __syncthreads();
  }
  if (t == 0) {
    const float cnt = sv[0];
    out[0] = (cnt > 0.0f) ? (sl[0] / cnt) : 0.0f;
  }
}

// ---------------------------------------------------------------------------
extern "C" void kernel_launch(void* const* d_in, const int* in_sizes, int n_in,
                              void* d_out, int out_size, void* d_ws, size_t ws_size,
                              hipStream_t stream) {
  (void)n_in; (void)out_size; (void)ws_size;
  const float* x      = (const float*)d_in[0];   // embeddings [B,128] fp32
  const int*   labels = (const int*)d_in[1];     // labels [B]
  float*       out    = (float*)d_out;           // scalar loss

  const int B = in_sizes[1];                     // 8192

  // Workspace layout (~4.1 MB):
  float* e         = (float*)d_ws;               // B*128 fp32 normalized
  float* sq        = e + (size_t)B * TL_D;       // B fp32 row norms^2
  float* row_loss  = sq + B;                     // B fp32
  float* row_valid = row_loss + B;               // B fp32

  tl_normalize<<<B / 8, 256, 0, stream>>>(x, e, sq);
  tl_mine<<<B / 128, 256, 0, stream>>>(e, sq, labels, row_loss, row_valid, B);
  tl_finalize<<<1, 256, 0, stream>>>(row_loss, row_valid, out, B);
}